// GumbelVectorQuantizer_61761629716883
// MI455X (gfx1250) — compile-verified
//
#include <hip/hip_runtime.h>

// ---------------------------------------------------------------------------
// Gumbel Vector Quantizer forward, fused, fp32 WMMA (gfx1250 / MI455X).
//
// GEMM: logits[32768,640] = x[32768,768] @ W[768,640] + b   (fp32 WMMA 16x16x4)
// Staging: double-buffered GLOBAL_LOAD_ASYNC_TO_LDS_B128 (ASYNCcnt pipelined),
// fallback to synchronous float4 copies if the async builtins are absent.
// Per row/group(320): argmax (hard counts), softmax sum (avg probs),
// gumbel argmax + y_soft@argmax -> out row = ((1+s)-s) * codebook[g,k*,:].
// Perplexities finalized by a tiny second kernel.
// ---------------------------------------------------------------------------

#define BT_TOTAL 32768
#define DDIM     768
#define GKDIM    640
#define KCW      320          // codewords per group
#define VD       128          // codebook dim per group
#define MT       64           // rows per workgroup
#define KC       32           // K chunk staged per iteration
#define NKCH     (DDIM / KC)  // 24
#define SA_STRIDE 36          // 32 + pad (16B aligned rows, conflict-free frag loads)
#define SB_STRIDE 648         // 640 + pad (16B aligned rows, disjoint banks for k,k+1)
#define SL_STRIDE 648
#define PBUF     (MT * SA_STRIDE + KC * SB_STRIDE)   // 23040 floats per stage buffer

#define AS1 __attribute__((address_space(1)))
#define AS3 __attribute__((address_space(3)))

#if defined(__AMDGCN__) && __has_builtin(__builtin_amdgcn_global_load_async_to_lds_b128) && \
    __has_builtin(__builtin_amdgcn_s_wait_asynccnt)
#define GVQ_ASYNC 1
#else
#define GVQ_ASYNC 0
#endif

typedef int v4i_ __attribute__((vector_size(16)));   // matches builtin param type
typedef __attribute__((ext_vector_type(2))) float v2f;
typedef __attribute__((ext_vector_type(8))) float v8f;

__global__ void gvq_zero_ws(float* __restrict__ w) {
    for (int i = threadIdx.x; i < 2 * GKDIM; i += 256) w[i] = 0.0f;
}

// Stage one K-chunk (64x32 A tile + 32x640 B tile) into LDS buffers dA/dB.
// Each thread issues exactly 22 transfers of 16 bytes.
__device__ __forceinline__ void stage_chunk(float* dA, float* dB,
                                            const float* __restrict__ X,
                                            const float* __restrict__ Wt,
                                            int rowBase, int kbase, int tid)
{
#pragma unroll
    for (int i = 0; i < 2; ++i) {
        int idx = tid + i * 256;
        int r   = idx >> 3;
        int c4  = (idx & 7) << 2;
        const float* g = X + (size_t)(rowBase + r) * DDIM + kbase + c4;
        float*       l = dA + r * SA_STRIDE + c4;
#if GVQ_ASYNC
        __builtin_amdgcn_global_load_async_to_lds_b128((AS1 v4i_*)g, (AS3 v4i_*)l, 0, 0);
#else
        *(float4*)l = *(const float4*)g;
#endif
    }
#pragma unroll 4
    for (int i = 0; i < 20; ++i) {
        int idx = tid + i * 256;
        int kr  = idx / 160;
        int c4  = (idx - kr * 160) << 2;
        const float* g = Wt + (size_t)(kbase + kr) * GKDIM + c4;
        float*       l = dB + kr * SB_STRIDE + c4;
#if GVQ_ASYNC
        __builtin_amdgcn_global_load_async_to_lds_b128((AS1 v4i_*)g, (AS3 v4i_*)l, 0, 0);
#else
        *(float4*)l = *(const float4*)g;
#endif
    }
}

__global__ __launch_bounds__(256, 1)
void gvq_main(const float* __restrict__ X,     // [32768,768]
              const float* __restrict__ Wt,    // [768,640]
              const float* __restrict__ Bias,  // [640]
              const float* __restrict__ Cb,    // [640,128]
              const float* __restrict__ U,     // [32768,640] uniform noise
              float* __restrict__ Out,         // [32768,256]
              float* __restrict__ gAvg,        // [640] softmax sums
              float* __restrict__ gCnt)        // [640] hard counts
{
    extern __shared__ float smem[];
    float* sAvg = smem;                         // 640 floats
    float* sReg = smem + GKDIM;                 // 2*PBUF floats (stage bufs) / logits union
    float* sL   = sReg;                         // [64][648]  (phase 2, aliases stage bufs)

    const int tid     = threadIdx.x;
    const int lane    = tid & 31;
    const int half    = lane & 15;
    const int wv      = tid >> 5;               // wave 0..7 -> N slice of 80
    const int rowBase = blockIdx.x * MT;

    for (int i = tid; i < GKDIM; i += 256) sAvg[i] = 0.0f;

    // ---------------- Phase 1: WMMA GEMM over K, double-buffered staging ----------------
    v8f acc[4][5];
#pragma unroll
    for (int mt = 0; mt < 4; ++mt)
#pragma unroll
        for (int nt = 0; nt < 5; ++nt)
#pragma unroll
            for (int e = 0; e < 8; ++e) acc[mt][nt][e] = 0.0f;

    // prologue: stage chunk 0 into buffer 0
    stage_chunk(sReg, sReg + MT * SA_STRIDE, X, Wt, rowBase, 0, tid);

    for (int kc = 0; kc < NKCH; ++kc) {
        // issue next chunk into the other buffer (it was fully consumed in the previous iteration)
        if (kc + 1 < NKCH) {
            float* nb = sReg + ((kc + 1) & 1) * PBUF;
            stage_chunk(nb, nb + MT * SA_STRIDE, X, Wt, rowBase, (kc + 1) * KC, tid);
#if GVQ_ASYNC
            __builtin_amdgcn_s_wait_asynccnt(22);   // oldest 22 (chunk kc) have landed
#endif
        } else {
#if GVQ_ASYNC
            __builtin_amdgcn_s_wait_asynccnt(0);
#endif
        }
        __syncthreads();                             // chunk kc visible to all waves

        const float* sA = sReg + (kc & 1) * PBUF;
        const float* sB = sA + MT * SA_STRIDE;
        const int klo = (lane >> 4) << 1;            // 0 (lanes 0-15) or 2 (lanes 16-31)
#pragma unroll
        for (int ks = 0; ks < 8; ++ks) {             // 8 steps of K=4
            const int kk = ks * 4 + klo;
            v2f a[4], bf[5];
#pragma unroll
            for (int mt = 0; mt < 4; ++mt) {         // A 16x4: lane=M, vgprs K{kk,kk+1}
                int m = mt * 16 + half;
                a[mt] = *(const v2f*)(&sA[m * SA_STRIDE + kk]);
            }
#pragma unroll
            for (int nt = 0; nt < 5; ++nt) {         // B 4x16: lane=N, vgprs K rows {kk,kk+1}
                int n = wv * 80 + nt * 16 + half;
                v2f b;
                b.x = sB[kk * SB_STRIDE + n];
                b.y = sB[(kk + 1) * SB_STRIDE + n];
                bf[nt] = b;
            }
#pragma unroll
            for (int mt = 0; mt < 4; ++mt)
#pragma unroll
                for (int nt = 0; nt < 5; ++nt)
                    acc[mt][nt] = __builtin_amdgcn_wmma_f32_16x16x4_f32(
                        false, a[mt], false, bf[nt], (short)0, acc[mt][nt], false, false);
        }
        __syncthreads();                             // done reading buf before it is re-staged
    }

    // ---------------- spill logits (+bias) to LDS (aliases staging) ----------------
    const int rowOff = (lane >> 4) << 3;             // C: vgpr e -> M = e (+8 for hi lanes)
#pragma unroll
    for (int mt = 0; mt < 4; ++mt)
#pragma unroll
        for (int nt = 0; nt < 5; ++nt) {
            int col = wv * 80 + nt * 16 + half;
            float bias = Bias[col];
#pragma unroll
            for (int e = 0; e < 8; ++e) {
                int row = mt * 16 + rowOff + e;
                sL[row * SL_STRIDE + col] = acc[mt][nt][e] + bias;
            }
        }
    __syncthreads();

    // ---------------- Phase 2: per (row, group) reductions ----------------
    const int sub = tid >> 4;                        // 16 subgroups of 16 lanes
    const int l16 = tid & 15;
    const int g   = sub & 1;                         // constant per thread across batches
    float accAvg[20];
#pragma unroll
    for (int j = 0; j < 20; ++j) accAvg[j] = 0.0f;

    for (int b = 0; b < 8; ++b) {
        const int rg  = b * 16 + sub;                // 128 row-groups: (row, g)
        const int row = rg >> 1;
        const int n   = rowBase + row;
        const float* Lr = &sL[row * SL_STRIDE + g * KCW];
        const float* Ur = U + (size_t)n * GKDIM + g * KCW;

        float m1 = -3.402823e38f; int k1 = 0;        // argmax logits (hard counts)
        float m2 = -3.402823e38f; int k2 = 0;        // argmax gumbel-perturbed
        float z[20];
#pragma unroll
        for (int j = 0; j < 20; ++j) {
            int k = l16 + j * 16;
            float lv = Lr[k];
            if (lv > m1) { m1 = lv; k1 = k; }
            float u  = Ur[k] * (1.0f - 2e-6f) + 1e-6f;
            float gn = -logf(-logf(u));
            float zz = (lv + gn) * 0.5f;             // / TAU (=2)
            z[j] = zz;
            if (zz > m2) { m2 = zz; k2 = k; }
        }
#pragma unroll
        for (int off = 8; off >= 1; off >>= 1) {
            float ov = __shfl_xor(m1, off, 16); int ok = __shfl_xor(k1, off, 16);
            if (ov > m1 || (ov == m1 && ok < k1)) { m1 = ov; k1 = ok; }
            float ow = __shfl_xor(m2, off, 16); int oj = __shfl_xor(k2, off, 16);
            if (ow > m2 || (ow == m2 && oj < k2)) { m2 = ow; k2 = oj; }
        }
        float e1a[20];
        float s1 = 0.0f, s2 = 0.0f;
#pragma unroll
        for (int j = 0; j < 20; ++j) {
            int k = l16 + j * 16;
            float e1 = expf(Lr[k] - m1);
            e1a[j] = e1; s1 += e1;
            s2 += expf(z[j] - m2);
        }
#pragma unroll
        for (int off = 8; off >= 1; off >>= 1) {
            s1 += __shfl_xor(s1, off, 16);
            s2 += __shfl_xor(s2, off, 16);
        }
        float inv1 = 1.0f / s1;
#pragma unroll
        for (int j = 0; j < 20; ++j) accAvg[j] += e1a[j] * inv1;

        // straight-through forward: y[k2] = (1 + s) - s, s = y_soft at its own argmax = 1/s2
        float s = 1.0f / s2;
        float f = (1.0f + s) - s;
        const float* cbr = Cb + (size_t)(g * KCW + k2) * VD;
        float* orow = Out + (size_t)n * (2 * VD) + g * VD;
#pragma unroll
        for (int j = 0; j < 8; ++j) {
            int d = l16 + j * 16;
            orow[d] = f * cbr[d];
        }
        if (l16 == 0) atomicAdd(&gCnt[g * KCW + k1], 1.0f);
    }

#pragma unroll
    for (int j = 0; j < 20; ++j)
        atomicAdd(&sAvg[g * KCW + l16 + j * 16], accAvg[j]);   // ds_add_f32
    __syncthreads();
    for (int i = tid; i < GKDIM; i += 256)
        atomicAdd(&gAvg[i], sAvg[i]);                          // one global atomic per WG
}

__global__ void gvq_finalize(const float* __restrict__ gAvg,
                             const float* __restrict__ gCnt,
                             float* __restrict__ outTail)      // [code_ppl, prob_ppl]
{
    const int t = threadIdx.x;                 // 32 threads (one wave)
    const float inv = 1.0f / (float)BT_TOTAL;
    float pa0 = 0.f, pa1 = 0.f, ph0 = 0.f, ph1 = 0.f;
    for (int k = t; k < KCW; k += 32) {
        float p0 = gAvg[k] * inv;          pa0 += p0 * logf(p0 + 1e-7f);
        float p1 = gAvg[KCW + k] * inv;    pa1 += p1 * logf(p1 + 1e-7f);
        float h0 = gCnt[k] * inv;          ph0 += h0 * logf(h0 + 1e-7f);
        float h1 = gCnt[KCW + k] * inv;    ph1 += h1 * logf(h1 + 1e-7f);
    }
#pragma unroll
    for (int off = 16; off >= 1; off >>= 1) {
        pa0 += __shfl_xor(pa0, off, 32);
        pa1 += __shfl_xor(pa1, off, 32);
        ph0 += __shfl_xor(ph0, off, 32);
        ph1 += __shfl_xor(ph1, off, 32);
    }
    if (t == 0) {
        outTail[0] = expf(-ph0) + expf(-ph1);   // code_perplexity (hard)
        outTail[1] = expf(-pa0) + expf(-pa1);   // prob_perplexity (soft)
    }
}

extern "C" void kernel_launch(void* const* d_in, const int* in_sizes, int n_in,
                              void* d_out, int out_size, void* d_ws, size_t ws_size,
                              hipStream_t stream) {
    (void)in_sizes; (void)n_in; (void)out_size; (void)ws_size;
    const float* X    = (const float*)d_in[0];
    const float* Wt   = (const float*)d_in[1];
    const float* Bias = (const float*)d_in[2];
    const float* Cb   = (const float*)d_in[3];
    const float* U    = (const float*)d_in[4];
    float* Out  = (float*)d_out;
    float* gAvg = (float*)d_ws;
    float* gCnt = gAvg + GKDIM;

    const size_t smemBytes = (size_t)(GKDIM + 2 * PBUF) * sizeof(float); // 186,880 B
    (void)hipFuncSetAttribute(reinterpret_cast<const void*>(gvq_main),
                              hipFuncAttributeMaxDynamicSharedMemorySize, (int)smemBytes);

    gvq_zero_ws<<<1, 256, 0, stream>>>(gAvg);
    gvq_main<<<BT_TOTAL / MT, 256, smemBytes, stream>>>(X, Wt, Bias, Cb, U, Out, gAvg, gCnt);
    gvq_finalize<<<1, 32, 0, stream>>>(gAvg, gCnt, Out + (size_t)BT_TOTAL * 2 * VD);
}